// model_gnn_89962384982332
// MI455X (gfx1250) — compile-verified
//
#include <hip/hip_runtime.h>
#include <math.h>

typedef __attribute__((ext_vector_type(2))) float v2f;
typedef __attribute__((ext_vector_type(4))) float v4f;
typedef __attribute__((ext_vector_type(8))) float v8f;

#define DIM 200
#define KR 8
#define CCOEF 0.85f
#define STRIPS 13    // ceil(200/16)
#define CHUNKS 50    // 200/4
#define LSTRIDE 212  // padded LDS row stride (floats): 16B-aligned, bank-conflict-free

__device__ __forceinline__ float softplus_f(float x) {
  return (x > 20.f) ? x : log1pf(__expf(x));
}

// One block per batch item. 256 threads = 8 wave32: waves 0-3 -> sign p, 4-7 -> sign n.
__global__ __launch_bounds__(256) void fused_gnn_kernel(
    const float* __restrict__ A,
    const float* __restrict__ kernel_p, const float* __restrict__ kernel_n,
    const float* __restrict__ Wp1, const float* __restrict__ Wp2, const float* __restrict__ Wp3,
    const float* __restrict__ Wn1, const float* __restrict__ Wn2, const float* __restrict__ Wn3,
    const float* __restrict__ lin1_w, const float* __restrict__ lin1_b,
    const float* __restrict__ lin2_w, const float* __restrict__ lin2_b,
    float* __restrict__ out)
{
  __shared__ float kpT[2][KR][LSTRIDE];   // relu(kernel), transposed   13.6 KB
  __shared__ v2f   bt[2][CHUNKS][32];     // WMMA B operand image       25.6 KB
  __shared__ float AFt[2][KR][LSTRIDE];   // (A @ kp)^T                 13.6 KB
  __shared__ float Rm[2][64];             // reduced 8x8
  __shared__ float Xa[2][64];             // gcn activations
  __shared__ float Tm[2][64];             // x @ W temp
  __shared__ float Ym[2][64];             // relu(I + c*AXW)
  __shared__ float hS[16];

  const int tid = threadIdx.x;
  const int b   = blockIdx.x;

  // ---- kp^T = relu(kernel)^T into LDS ----
  for (int i = tid; i < 2 * DIM * KR; i += 256) {
    int sign = (i < DIM * KR) ? 0 : 1;
    int j = i - sign * DIM * KR;         // j = row*8 + col of kernel
    int row = j / KR, col = j % KR;
    const float* Kk = sign ? kernel_n : kernel_p;
    kpT[sign][col][row] = fmaxf(Kk[j], 0.f);
  }
  __syncthreads();

  // ---- Precompute per-lane WMMA B operand (4x16 kp tile, cols 8..15 zero-padded) ----
  // B layout: VGPR0 = rows {k0, k0+2} (lo/hi half-wave), VGPR1 = rows {k0+1, k0+3}, N = lane&15
  for (int i = tid; i < 2 * CHUNKS * 32; i += 256) {
    int sign = i / (CHUNKS * 32);
    int rem  = i % (CHUNKS * 32);
    int c    = rem >> 5;
    int lane = rem & 31;
    int col  = lane & 15;
    int row  = 4 * c + ((lane >> 4) << 1);
    v2f v;
    if (col < KR) { v.x = kpT[sign][col][row]; v.y = kpT[sign][col][row + 1]; }
    else          { v.x = 0.f;                 v.y = 0.f; }
    bt[sign][c][lane] = v;
  }
  __syncthreads();

  // ---- Stage 1: AF^T = (A @ kp)^T via V_WMMA_F32_16X16X4_F32, streaming A once ----
  {
    const int wave = tid >> 5;
    const int lane = tid & 31;
    const int sign = wave >> 2;      // 0 = p, 1 = n
    const int wq   = wave & 3;       // strip round-robin among 4 waves
    const float* Ab = A + ((size_t)b * 2 + sign) * (DIM * DIM);
    const int rl     = lane & 15;
    const int colofs = (lane >> 4) << 1;   // hi half-wave handles K+2,K+3

    for (int s = wq; s < STRIPS; s += 4) {
      int row = s * 16 + rl;
      if (row > DIM - 1) row = DIM - 1;    // clamp: results land in AFt cols >= 200, never read
      const float* ap = Ab + row * DIM + colofs;
      v8f acc = {0.f, 0.f, 0.f, 0.f, 0.f, 0.f, 0.f, 0.f};
      #pragma unroll 5
      for (int c = 0; c < CHUNKS; ++c) {
        v2f av = *(const v2f*)(ap + c * 4);   // A-operand: lane-exact layout
        v2f bv = bt[sign][c][lane];           // B-operand from LDS
        acc = __builtin_amdgcn_wmma_f32_16x16x4_f32(
            false, av, false, bv, (short)0, acc, false, false);
      }
      // D layout: VGPR v -> strip row (v + 8*hiHalf), col n = lane&15.
      // Transposed store: AFt[sign][n][s*16 + mbase + v] -> two contiguous b128 stores.
      if (rl < KR) {
        const int gbase = s * 16 + ((lane >> 4) << 3);
        float* dst = &AFt[sign][rl][gbase];
        v4f lo = {acc[0], acc[1], acc[2], acc[3]};
        v4f hi = {acc[4], acc[5], acc[6], acc[7]};
        *(v4f*)(dst)     = lo;
        *(v4f*)(dst + 4) = hi;
      }
    }
  }
  __syncthreads();

  // ---- Stage 2: reduced R = kp^T @ AF  (8x8 per sign), vectorized along i ----
  if (tid < 128) {
    int sgn = tid >> 6, idx = tid & 63, a = idx >> 3, m = idx & 7;
    const v4f* kp4 = (const v4f*)&kpT[sgn][a][0];
    const v4f* af4 = (const v4f*)&AFt[sgn][m][0];
    float s = 0.f;
    #pragma unroll 5
    for (int i = 0; i < DIM / 4; ++i) {
      v4f x = kp4[i], y = af4[i];
      s += x.x * y.x + x.y * y.y + x.z * y.z + x.w * y.w;
    }
    Rm[sgn][idx] = s;
  }
  __syncthreads();

  // ---- GCN chain (8x8) ----
  const bool act = (tid < 128);
  const int sgn = (tid >> 6) & 1;
  const int idx = tid & 63;
  const int r = idx >> 3, c = idx & 7;
  const float* WL[2][3] = { {Wp1, Wp2, Wp3}, {Wn1, Wn2, Wn3} };

  // layer 1: AXW = R @ W
  {
    const float* W = WL[sgn][0];
    float y = 0.f;
    if (act) {
      float axw = 0.f;
      #pragma unroll
      for (int k = 0; k < 8; ++k) axw += Rm[sgn][r * 8 + k] * W[k * 8 + c];
      y = ((r == c) ? 1.f : 0.f) + CCOEF * axw;
      y = fmaxf(y, 0.f);
      Ym[sgn][idx] = y;
    }
    __syncthreads();
    if (act) {
      float cm = 0.f;
      #pragma unroll
      for (int k = 0; k < 8; ++k) cm += Ym[sgn][k * 8 + c];
      cm = cm * 0.125f + 1e-6f;
      Xa[sgn][idx] = softplus_f(y / cm);
    }
    __syncthreads();
  }
  // layers 2,3: AXW = R @ (X @ W)
  for (int L = 1; L < 3; ++L) {
    const float* W = WL[sgn][L];
    if (act) {
      float xw = 0.f;
      #pragma unroll
      for (int k = 0; k < 8; ++k) xw += Xa[sgn][r * 8 + k] * W[k * 8 + c];
      Tm[sgn][idx] = xw;
    }
    __syncthreads();
    float y = 0.f;
    if (act) {
      float axw = 0.f;
      #pragma unroll
      for (int k = 0; k < 8; ++k) axw += Rm[sgn][r * 8 + k] * Tm[sgn][k * 8 + c];
      y = ((r == c) ? 1.f : 0.f) + CCOEF * axw;
      y = fmaxf(y, 0.f);
      Ym[sgn][idx] = y;
    }
    __syncthreads();
    if (act) {
      float cm = 0.f;
      #pragma unroll
      for (int k = 0; k < 8; ++k) cm += Ym[sgn][k * 8 + c];
      cm = cm * 0.125f + 1e-6f;
      Xa[sgn][idx] = softplus_f(y / cm);
    }
    __syncthreads();
  }

  // ---- Linear head: concat([p3, n3], -1).reshape(128) @ lin1^T + b1 -> @ lin2^T + b2 ----
  if (tid < 16) {
    float h = lin1_b[tid];
    for (int k = 0; k < 128; ++k) {
      int row = k >> 4, col = k & 15;
      float v = (col < 8) ? Xa[0][row * 8 + col] : Xa[1][row * 8 + (col - 8)];
      h += v * lin1_w[tid * 128 + k];
    }
    hS[tid] = h;
  }
  __syncthreads();
  if (tid < 2) {
    float o = lin2_b[tid];
    #pragma unroll
    for (int j = 0; j < 16; ++j) o += hS[j] * lin2_w[tid * 16 + j];
    out[(size_t)b * 2 + tid] = o;
  }
}

// Batch-independent scalar loss (one tiny block).
__global__ __launch_bounds__(256) void loss_kernel(
    const float* __restrict__ kernel_p, const float* __restrict__ kernel_n,
    const float* __restrict__ Wp1, const float* __restrict__ Wp2, const float* __restrict__ Wp3,
    const float* __restrict__ Wn1, const float* __restrict__ Wn2, const float* __restrict__ Wn3,
    float* __restrict__ loss_out)
{
  __shared__ float gram[2][64];
  __shared__ float red[256];
  const int tid = threadIdx.x;
  float part = 0.f;

  for (int sign = 0; sign < 2; ++sign) {
    const float* K = sign ? kernel_n : kernel_p;
    for (int i = tid; i < DIM * KR; i += 256) {
      float v = K[i];
      part += 0.1f  * fmaxf(1e-6f - v, 0.f);  // neg penalty on kernel
      part += 0.05f * fmaxf(v, 0.f);          // L1 of relu(kernel)
    }
    if (tid < 64) {
      int a = tid >> 3, bc = tid & 7;
      float s = 0.f;
      for (int i = 0; i < DIM; ++i)
        s += fmaxf(K[i * KR + a], 0.f) * fmaxf(K[i * KR + bc], 0.f);
      gram[sign][tid] = s;
    }
  }
  {
    const float* Ws[6] = {Wp1, Wp2, Wp3, Wn1, Wn2, Wn3};
    for (int w = 0; w < 6; ++w)
      if (tid < 64) part += 0.2f * fmaxf(1e-6f - Ws[w][tid], 0.f);
  }
  red[tid] = part;
  __syncthreads();
  for (int s = 128; s > 0; s >>= 1) {
    if (tid < s) red[tid] += red[tid + s];
    __syncthreads();
  }
  if (tid == 0) {
    float tot = red[0];
    for (int sign = 0; sign < 2; ++sign) {
      float vp = sign ? 0.5f : 0.3f;
      float dsum = 0.f;
      for (int a = 0; a < 8; ++a) dsum += gram[sign][a * 9];
      float mean = dsum * 0.125f;
      float var = 0.f;
      for (int a = 0; a < 8; ++a) { float d = gram[sign][a * 9] - mean; var += d * d; }
      var *= (1.f / 7.f);  // ddof=1
      float orth = 0.f;
      for (int i = 0; i < 64; ++i) {
        if ((i >> 3) != (i & 7)) { float g = gram[sign][i]; orth += g * g; }
      }
      tot += 0.2f * orth + vp * var;
    }
    loss_out[0] = tot;
  }
}

extern "C" void kernel_launch(void* const* d_in, const int* in_sizes, int n_in,
                              void* d_out, int out_size, void* d_ws, size_t ws_size,
                              hipStream_t stream) {
  const float* A        = (const float*)d_in[0];
  const float* kernel_p = (const float*)d_in[1];
  const float* kernel_n = (const float*)d_in[2];
  const float* Wp1 = (const float*)d_in[3];
  const float* Wp2 = (const float*)d_in[4];
  const float* Wp3 = (const float*)d_in[5];
  const float* Wn1 = (const float*)d_in[6];
  const float* Wn2 = (const float*)d_in[7];
  const float* Wn3 = (const float*)d_in[8];
  const float* lin1_w = (const float*)d_in[9];
  const float* lin1_b = (const float*)d_in[10];
  const float* lin2_w = (const float*)d_in[11];
  const float* lin2_b = (const float*)d_in[12];
  float* out = (float*)d_out;

  const int B = in_sizes[0] / (2 * DIM * DIM);  // 4096

  fused_gnn_kernel<<<B, 256, 0, stream>>>(
      A, kernel_p, kernel_n, Wp1, Wp2, Wp3, Wn1, Wn2, Wn3,
      lin1_w, lin1_b, lin2_w, lin2_b, out);

  loss_kernel<<<1, 256, 0, stream>>>(
      kernel_p, kernel_n, Wp1, Wp2, Wp3, Wn1, Wn2, Wn3,
      out + (size_t)out_size - 1);
}